// MpGNN_83983790506199
// MI455X (gfx1250) — compile-verified
//
#include <hip/hip_runtime.h>
#include <hip/hip_bf16.h>

#define FD 64   // feature dim

typedef float v2f __attribute__((ext_vector_type(2)));
typedef float v8f __attribute__((ext_vector_type(8)));

// ---------------------------------------------------------------------------
// Zero-fill (grid-stride). Needed because scatter/pool buffers are accumulated
// with atomics and the harness does not re-zero workspace between replays.
// ---------------------------------------------------------------------------
__global__ void mpgnn_zero_kernel(float* __restrict__ p, long long n) {
  long long i = (long long)blockIdx.x * blockDim.x + threadIdx.x;
  long long stride = (long long)gridDim.x * blockDim.x;
  for (; i < n; i += stride) p[i] = 0.0f;
}

// ---------------------------------------------------------------------------
// Edge scatter: agg[dst] += x[src] * ew.  16 lanes per edge, 4 dims per lane.
// Gather is a coalesced B128 (float4); scatter is 4 coalesced f32 atomics.
// Whole x (12.8 MB) is L2-resident on MI455X (192 MB L2).
// ---------------------------------------------------------------------------
__global__ void mpgnn_edge_scatter_kernel(const float* __restrict__ xin,
                                          const int* __restrict__ src,
                                          const int* __restrict__ dst,
                                          const float* __restrict__ ew,
                                          float* __restrict__ agg,
                                          int nEdges) {
  int t = blockIdx.x * blockDim.x + threadIdx.x;
  int e = t >> 4;
  if (e >= nEdges) return;
  int q  = (t & 15) << 2;          // dim base (0..60)
  int s  = src[e];
  int d0 = dst[e];
  float w = ew[e];
  const float4 v = *reinterpret_cast<const float4*>(xin + (size_t)s * FD + q);
  float* o = agg + (size_t)d0 * FD + q;
  atomicAdd(o + 0, v.x * w);
  atomicAdd(o + 1, v.y * w);
  atomicAdd(o + 2, v.z * w);
  atomicAdd(o + 3, v.w * w);
}

// ---------------------------------------------------------------------------
// Fused dual GEMM: out = [relu]( A@Wrel + X@Wroot + bias ), optionally
// atomically pooled into per-graph sums instead of storing.
//
// Both 64x64 weight matrices are staged in LDS once per block (coalesced
// B128 fill), so the inner loop's B-fragments come from ds_load instead of
// per-lane strided global loads.  One wave per 16-node row tile; 4 column
// tiles of 16; K swept in steps of 4 with V_WMMA_F32_16X16X4_F32 (exact f32
// math on the matrix pipes).
//
// Fragment layouts (wave32, lane L, h = L/16, m = L%16):
//   A (16x4 f32):  a = { A[m][k+2h], A[m][k+2h+1] }        (aligned float2)
//   B (4x16 f32):  b = { W[k+2h][n], W[k+2h+1][n] },  n = ntile*16 + m
//   C/D (16x16):   acc[v] = D[v + 8h][n]
// All branches around the WMMAs are wave-uniform -> EXEC is all ones.
// ---------------------------------------------------------------------------
__global__ void mpgnn_dualgemm_kernel(const float* __restrict__ A0,
                                      const float* __restrict__ X0,
                                      const float* __restrict__ Wrel,
                                      const float* __restrict__ Wroot,
                                      const float* __restrict__ bias,
                                      float* __restrict__ out,     // may be null
                                      int nRows, int doRelu,
                                      const int* __restrict__ batch, // null unless pooling
                                      float* __restrict__ pool) {   // [G,64] sums
  __shared__ float sWrel[FD * FD];    // 16 KB
  __shared__ float sWroot[FD * FD];   // 16 KB  (32 KB total < 320 KB/WGP)

  // Cooperative coalesced fill: 256 threads x 4 iters x float4 per matrix.
  for (int i = threadIdx.x * 4; i < FD * FD; i += blockDim.x * 4) {
    *reinterpret_cast<float4*>(&sWrel[i]) =
        *reinterpret_cast<const float4*>(&Wrel[i]);
    *reinterpret_cast<float4*>(&sWroot[i]) =
        *reinterpret_cast<const float4*>(&Wroot[i]);
  }
  __syncthreads();

  int wave = blockIdx.x * (blockDim.x >> 5) + (threadIdx.x >> 5);
  int lane = threadIdx.x & 31;
  int row0 = wave * 16;
  if (row0 >= nRows) return;            // wave-uniform exit (after barrier)

  int m = lane & 15;
  int h = lane >> 4;                    // 0 or 1
  size_t rowA = (size_t)(row0 + m) * FD;

  for (int nt = 0; nt < 4; ++nt) {
    int ncol = nt * 16 + m;             // output column for this lane
    float bv = bias[ncol];
    v8f acc;
#pragma unroll
    for (int v = 0; v < 8; ++v) acc[v] = bv;   // bias folded into accumulator

#pragma unroll
    for (int k = 0; k < FD; k += 4) {
      int ka = k + 2 * h;
      v2f a = *reinterpret_cast<const v2f*>(A0 + rowA + ka);
      v2f b;
      b[0] = sWrel[ka * FD + ncol];
      b[1] = sWrel[(ka + 1) * FD + ncol];
      acc = __builtin_amdgcn_wmma_f32_16x16x4_f32(
          false, a, false, b, (short)0, acc, false, false);

      v2f a2 = *reinterpret_cast<const v2f*>(X0 + rowA + ka);
      v2f b2;
      b2[0] = sWroot[ka * FD + ncol];
      b2[1] = sWroot[(ka + 1) * FD + ncol];
      acc = __builtin_amdgcn_wmma_f32_16x16x4_f32(
          false, a2, false, b2, (short)0, acc, false, false);
    }

#pragma unroll
    for (int v = 0; v < 8; ++v) {
      int rr = row0 + v + 8 * h;        // D-matrix row for this VGPR
      float val = acc[v];
      if (doRelu) val = fmaxf(val, 0.0f);
      if (out)  out[(size_t)rr * FD + ncol] = val;
      if (pool) atomicAdd(&pool[(size_t)batch[rr] * FD + ncol], val);
    }
  }
}

// ---------------------------------------------------------------------------
// Per-graph node counts for mean pooling.
// ---------------------------------------------------------------------------
__global__ void mpgnn_count_kernel(const int* __restrict__ batch,
                                   float* __restrict__ cnt, int n) {
  int i = blockIdx.x * blockDim.x + threadIdx.x;
  if (i < n) atomicAdd(&cnt[batch[i]], 1.0f);
}

// ---------------------------------------------------------------------------
// Head: out[g][c] = sigmoid( (sums[g]/max(cnt,1)) . Wfc[:,c] + bfc[c] ).
// 1024 outputs -> trivially cheap, plain VALU.
// ---------------------------------------------------------------------------
__global__ void mpgnn_head_kernel(const float* __restrict__ sums,
                                  const float* __restrict__ cnt,
                                  const float* __restrict__ Wfc,  // [64,2]
                                  const float* __restrict__ bfc,  // [2]
                                  float* __restrict__ outp, int G) {
  int t = blockIdx.x * blockDim.x + threadIdx.x;
  if (t >= G * 2) return;
  int g = t >> 1, c = t & 1;
  float inv = 1.0f / fmaxf(cnt[g], 1.0f);
  float acc = bfc[c];
#pragma unroll
  for (int d = 0; d < FD; ++d)
    acc += sums[(size_t)g * FD + d] * inv * Wfc[d * 2 + c];
  outp[t] = 1.0f / (1.0f + expf(-acc));
}

// ---------------------------------------------------------------------------
extern "C" void kernel_launch(void* const* d_in, const int* in_sizes, int n_in,
                              void* d_out, int out_size, void* d_ws, size_t ws_size,
                              hipStream_t stream) {
  const float* x     = (const float*)d_in[0];
  const int*   ei    = (const int*)  d_in[1];
  const float* ea    = (const float*)d_in[2];
  const int*   batch = (const int*)  d_in[3];
  const float* W1r   = (const float*)d_in[4];
  const float* b1    = (const float*)d_in[5];
  const float* W1o   = (const float*)d_in[6];
  const float* W2r   = (const float*)d_in[7];
  const float* b2    = (const float*)d_in[8];
  const float* W2o   = (const float*)d_in[9];
  const float* Wfc   = (const float*)d_in[10];
  const float* bfc   = (const float*)d_in[11];
  float* out = (float*)d_out;

  const int N = in_sizes[0] / FD;      // 50000
  const int E = in_sizes[2];           // 800000
  const int G = out_size / 2;          // 512
  const int* srcI = ei;                // edge_index[0]
  const int* dstI = ei + E;            // edge_index[1]

  char* ws = (char*)d_ws;
  size_t nodeBytes = (size_t)N * FD * sizeof(float);
  float* agg  = (float*)ws;                    // reused for both layers
  float* h1   = (float*)(ws + nodeBytes);
  float* sums = (float*)(ws + 2 * nodeBytes);  // [G,64]
  float* cnt  = sums + (size_t)G * FD;         // [G]

  // ---- layer 1 ----
  mpgnn_zero_kernel<<<1024, 256, 0, stream>>>(agg, (long long)N * FD);
  mpgnn_zero_kernel<<<64, 256, 0, stream>>>(sums, (long long)G * FD + G);

  int sThreads = E * 16;
  mpgnn_edge_scatter_kernel<<<(sThreads + 255) / 256, 256, 0, stream>>>(
      x, srcI, dstI, ea, agg, E);

  int tiles = (N + 15) / 16;
  int gBlocks = (tiles + 7) / 8;               // 8 waves (tiles) per block
  mpgnn_dualgemm_kernel<<<gBlocks, 256, 0, stream>>>(
      agg, x, W1r, W1o, b1, h1, N, /*relu=*/1, nullptr, nullptr);

  // ---- layer 2 (pooling fused, h2 never materialized) ----
  mpgnn_zero_kernel<<<1024, 256, 0, stream>>>(agg, (long long)N * FD);
  mpgnn_edge_scatter_kernel<<<(sThreads + 255) / 256, 256, 0, stream>>>(
      h1, srcI, dstI, ea, agg, E);
  mpgnn_count_kernel<<<(N + 255) / 256, 256, 0, stream>>>(batch, cnt, N);
  mpgnn_dualgemm_kernel<<<gBlocks, 256, 0, stream>>>(
      agg, h1, W2r, W2o, b2, nullptr, N, /*relu=*/0, batch, sums);

  // ---- head ----
  mpgnn_head_kernel<<<(G * 2 + 255) / 256, 256, 0, stream>>>(
      sums, cnt, Wfc, bfc, out, G);
}